// VisionEncoderMambaBlock3_74345883893856
// MI455X (gfx1250) — compile-verified
//
#include <hip/hip_runtime.h>
#include <hip/hip_bf16.h>
#include <math.h>

// ---------------------------------------------------------------------------
// Mamba-style vision block for MI455X (gfx1250, wave32).
//  - GEMMs: V_WMMA_F32_16X16X4_F32, one wave -> 16x64 tile (A-frag reuse x4).
//  - Selective scan: exact chunked linear-recurrence decomposition
//    (32 chunks x 64 steps) -> 32x shorter critical path, 1024+ live waves.
// ---------------------------------------------------------------------------

typedef float v2f __attribute__((ext_vector_type(2)));
typedef float v8f __attribute__((ext_vector_type(8)));

#define BATCH  2
#define SEQ    2048
#define DIM    512
#define DTR    32
#define DST    16
#define ROWS   (BATCH * SEQ)   // 4096 tokens per branch
#define NCHUNK 32
#define CLEN   (SEQ / NCHUNK)  // 64

// --------------------------- LayerNorm -------------------------------------
__global__ __launch_bounds__(256)
void ln_kernel(const float* __restrict__ X, const float* __restrict__ w,
               const float* __restrict__ b, float* __restrict__ out, int rows) {
  const int wave = threadIdx.x >> 5;
  const int lane = threadIdx.x & 31;
  const int row  = blockIdx.x * (blockDim.x >> 5) + wave;
  if (row >= rows) return;
  const float* xr = X + (size_t)row * DIM;
  float v[16];
  float s = 0.f, s2 = 0.f;
#pragma unroll
  for (int j = 0; j < 16; ++j) {
    float t = xr[lane + 32 * j];
    v[j] = t; s += t; s2 += t * t;
  }
#pragma unroll
  for (int o = 16; o >= 1; o >>= 1) {
    s  += __shfl_xor(s,  o, 32);
    s2 += __shfl_xor(s2, o, 32);
  }
  const float mu  = s * (1.f / DIM);
  const float var = s2 * (1.f / DIM) - mu * mu;
  const float inv = rsqrtf(var + 1e-5f);
  float* orow = out + (size_t)row * DIM;
#pragma unroll
  for (int j = 0; j < 16; ++j) {
    const int idx = lane + 32 * j;
    orow[idx] = (v[j] - mu) * inv * w[idx] + b[idx];
  }
}

// --------------------------- WMMA f32 GEMM ---------------------------------
// C[M x N] = A[M x K] * W[N x K]^T (+ bias). One wave -> 16x64 tile:
// 4 accumulators share each A fragment (load:wmma = 5:4).
// mode 0: plain   mode 1: also Z = silu(C)   mode 2: C = softplus(C)
__global__ __launch_bounds__(32)
void gemm_wmma_f32(const float* __restrict__ A, const float* __restrict__ W,
                   const float* __restrict__ bias, float* __restrict__ C,
                   float* __restrict__ Z, int N, int K, int lda, int ldw,
                   int mode) {
  const int lane = threadIdx.x & 31;
  const int n0 = blockIdx.x * 64;
  const int m0 = blockIdx.y * 16;
  const int ml   = lane & 15;          // row (A) / col (B) within sub-tile
  const int koff = (lane >> 4) << 1;   // lane-half K offset per ISA layout

  v8f acc[4];
#pragma unroll
  for (int j = 0; j < 4; ++j)
#pragma unroll
    for (int i = 0; i < 8; ++i) acc[j][i] = 0.f;

  const float* arow = A + (size_t)(m0 + ml) * lda + koff;
  const float* brow = W + (size_t)(n0 + ml) * ldw + koff;

#pragma unroll 2
  for (int k = 0; k < K; k += 4) {
    v2f a;
    a[0] = arow[k]; a[1] = arow[k + 1];
#pragma unroll
    for (int j = 0; j < 4; ++j) {
      const size_t bo = (size_t)j * 16 * ldw + k;
      v2f b;
      b[0] = brow[bo]; b[1] = brow[bo + 1];
      acc[j] = __builtin_amdgcn_wmma_f32_16x16x4_f32(
          false, a, false, b, (short)0, acc[j], false, false);
    }
  }

  const int rbase = m0 + ((lane >> 4) << 3);  // lanes 16-31 hold M=i+8
#pragma unroll
  for (int j = 0; j < 4; ++j) {
    const int col  = n0 + 16 * j + ml;
    const float bs = bias ? bias[col] : 0.f;
#pragma unroll
    for (int i = 0; i < 8; ++i) {
      float val = acc[j][i] + bs;
      const size_t off = (size_t)(rbase + i) * N + col;
      if (mode == 1) {
        C[off] = val;
        Z[off] = val / (1.f + __expf(-val));                // silu
      } else if (mode == 2) {
        C[off] = (val > 30.f) ? val : log1pf(__expf(val));  // softplus
      } else {
        C[off] = val;
      }
    }
  }
}

// --------------------- Selective scan: chunked (exact) ----------------------
// h_l = dA_l * h_{l-1} + dBu_l is linear: decompose into NCHUNK chunks.
// Pass 1: per (b,d,n,chunk) compute (prod dA, h_end from h0=0).
// Pass 2: serial scan over NCHUNK summaries -> per-chunk initial state.
// Pass 3: replay chunk from its initial state and emit y.

__global__ __launch_bounds__(32)
void scan_pass1(const float* __restrict__ u_buf,
                const float* __restrict__ delta_buf,
                const float* __restrict__ dbc,
                const float* __restrict__ A_log,
                float* __restrict__ aprod_buf,   // [B*DIM*DST*NCHUNK]
                float* __restrict__ hend_buf) {  // [B*DIM*DST*NCHUNK]
  const int lane = threadIdx.x & 31;
  const int d  = blockIdx.x * 32 + lane;
  const int bb = blockIdx.y;
  const int c  = blockIdx.z;

  float a[DST], h[DST], ap[DST];
#pragma unroll
  for (int n = 0; n < DST; ++n) {
    a[n]  = -__expf(A_log[(size_t)d * DST + n]);
    h[n]  = 0.f;
    ap[n] = 1.f;
  }

  const int l0 = c * CLEN;
  for (int l = l0; l < l0 + CLEN; ++l) {
    const size_t t = (size_t)bb * SEQ + l;
    const float dlt = delta_buf[t * DIM + d];
    const float du  = dlt * u_buf[t * DIM + d];
    const float4* bc = (const float4*)(dbc + t * 64 + DTR);
    float4 Bq[4];
#pragma unroll
    for (int q = 0; q < 4; ++q) Bq[q] = bc[q];
    const float* Bv = (const float*)Bq;
#pragma unroll
    for (int n = 0; n < DST; ++n) {
      const float dA = __expf(dlt * a[n]);
      ap[n] = ap[n] * dA;
      h[n]  = dA * h[n] + du * Bv[n];
    }
  }

  const size_t base = (((size_t)bb * DIM + d) * DST) * NCHUNK + c;
#pragma unroll
  for (int n = 0; n < DST; ++n) {
    aprod_buf[base + (size_t)n * NCHUNK] = ap[n];
    hend_buf [base + (size_t)n * NCHUNK] = h[n];
  }
}

// One thread per (b,d,n): serial scan over NCHUNK summaries; overwrite
// hend_buf with each chunk's *initial* state (exclusive scan).
__global__ __launch_bounds__(256)
void scan_pass2(const float* __restrict__ aprod_buf,
                float* __restrict__ hend_buf) {
  const int idx = blockIdx.x * blockDim.x + threadIdx.x;  // (b,d,n) flat
  if (idx >= BATCH * DIM * DST) return;
  const size_t base = (size_t)idx * NCHUNK;
  float h = 0.f;
#pragma unroll
  for (int c = 0; c < NCHUNK; ++c) {
    const float apc = aprod_buf[base + c];
    const float hec = hend_buf[base + c];
    hend_buf[base + c] = h;        // initial state for chunk c
    h = apc * h + hec;             // state after chunk c
  }
}

__global__ __launch_bounds__(32)
void scan_pass3(const float* __restrict__ u_buf,
                const float* __restrict__ delta_buf,
                const float* __restrict__ dbc,
                const float* __restrict__ A_log,
                const float* __restrict__ Dv,
                const float* __restrict__ hinit_buf,  // = hend_buf after pass2
                float* __restrict__ y_buf) {
  const int lane = threadIdx.x & 31;
  const int d  = blockIdx.x * 32 + lane;
  const int bb = blockIdx.y;
  const int c  = blockIdx.z;

  const size_t base = (((size_t)bb * DIM + d) * DST) * NCHUNK + c;
  float a[DST], h[DST];
#pragma unroll
  for (int n = 0; n < DST; ++n) {
    a[n] = -__expf(A_log[(size_t)d * DST + n]);
    h[n] = hinit_buf[base + (size_t)n * NCHUNK];
  }
  const float Dd = Dv[d];

  const int l0 = c * CLEN;
  for (int l = l0; l < l0 + CLEN; ++l) {
    const size_t t = (size_t)bb * SEQ + l;
    const float dlt = delta_buf[t * DIM + d];
    const float u   = u_buf[t * DIM + d];
    const float du  = dlt * u;
    const float4* bc = (const float4*)(dbc + t * 64 + DTR);
    float4 Bq[4], Cq[4];
#pragma unroll
    for (int q = 0; q < 4; ++q) { Bq[q] = bc[q]; Cq[q] = bc[4 + q]; }
    const float* Bv = (const float*)Bq;
    const float* Cv = (const float*)Cq;
    float y = 0.f;
#pragma unroll
    for (int n = 0; n < DST; ++n) {
      const float dA = __expf(dlt * a[n]);
      h[n] = dA * h[n] + du * Bv[n];
      y += h[n] * Cv[n];
    }
    y_buf[t * DIM + d] = y + u * Dd;
  }
}

// --------------------------- Combine ---------------------------------------
// out = y0*(z1+z2) + y1*(z0+z2) + 2*y2*z0
__global__ __launch_bounds__(256)
void combine_kernel(const float* __restrict__ y0, const float* __restrict__ y1,
                    const float* __restrict__ y2, const float* __restrict__ z0,
                    const float* __restrict__ z1, const float* __restrict__ z2,
                    float* __restrict__ out, int n) {
  const int i = blockIdx.x * blockDim.x + threadIdx.x;
  if (i >= n) return;
  out[i] = y0[i] * (z1[i] + z2[i]) + y1[i] * (z0[i] + z2[i]) +
           2.f * y2[i] * z0[i];
}

// ---------------------------------------------------------------------------
extern "C" void kernel_launch(void* const* d_in, const int* in_sizes, int n_in,
                              void* d_out, int out_size, void* d_ws,
                              size_t ws_size, hipStream_t stream) {
  const float* X[3] = {(const float*)d_in[0], (const float*)d_in[1],
                       (const float*)d_in[2]};
  const float* norm_w    = (const float*)d_in[3];
  const float* norm_b    = (const float*)d_in[4];
  const float* proj_w    = (const float*)d_in[5];
  const float* proj_b    = (const float*)d_in[6];
  const float* conv_w    = (const float*)d_in[7];
  const float* conv_b    = (const float*)d_in[8];
  const float* deltaBC_w = (const float*)d_in[9];
  const float* dt_w      = (const float*)d_in[10];
  const float* dt_b      = (const float*)d_in[11];
  const float* A_log     = (const float*)d_in[12];
  const float* Dv        = (const float*)d_in[13];

  constexpr size_t MT = (size_t)ROWS * DIM;          // 2,097,152 floats
  constexpr size_t CS = (size_t)BATCH * DIM * DST * NCHUNK;  // 524,288 floats
  float* ws    = (float*)d_ws;
  float* tn    = ws;             // reused as delta after proj GEMM consumes it
  float* p     = ws + MT;        // reused as dBC after conv GEMM consumes it
  float* conv  = ws + 2 * MT;
  float* z     = ws + 3 * MT;    // z[3]
  float* y     = ws + 6 * MT;    // y[3]
  float* aprod = ws + 9 * MT;    // chunk dA products
  float* hend  = aprod + CS;     // chunk end states -> initial states
  float* delta = tn;
  float* dbc   = p;              // [ROWS x 64]

  for (int bb = 0; bb < 3; ++bb) {
    // 1) LayerNorm
    ln_kernel<<<ROWS / 8, 256, 0, stream>>>(X[bb], norm_w, norm_b, tn, ROWS);

    // 2) p = tn @ proj_w^T + proj_b ; z = silu(p)
    gemm_wmma_f32<<<dim3(DIM / 64, ROWS / 16), 32, 0, stream>>>(
        tn, proj_w, proj_b, p, z + (size_t)bb * MT, DIM, DIM, DIM, DIM, 1);

    // 3) conv = p @ conv_w^T + conv_b
    gemm_wmma_f32<<<dim3(DIM / 64, ROWS / 16), 32, 0, stream>>>(
        p, conv_w, conv_b, conv, nullptr, DIM, DIM, DIM, DIM, 0);

    // 4) dBC = conv @ deltaBC_w^T        (N = 64, no bias)
    gemm_wmma_f32<<<dim3(1, ROWS / 16), 32, 0, stream>>>(
        conv, deltaBC_w, nullptr, dbc, nullptr, 64, DIM, DIM, DIM, 0);

    // 5) delta = softplus(dBC[:, :32] @ dt_w^T + dt_b)   (K = 32)
    gemm_wmma_f32<<<dim3(DIM / 64, ROWS / 16), 32, 0, stream>>>(
        dbc, dt_w, dt_b, delta, nullptr, DIM, DTR, 64, DTR, 2);

    // 6) chunked selective scan (exact linear-recurrence decomposition)
    scan_pass1<<<dim3(DIM / 32, BATCH, NCHUNK), 32, 0, stream>>>(
        conv, delta, dbc, A_log, aprod, hend);
    scan_pass2<<<(BATCH * DIM * DST) / 256, 256, 0, stream>>>(aprod, hend);
    scan_pass3<<<dim3(DIM / 32, BATCH, NCHUNK), 32, 0, stream>>>(
        conv, delta, dbc, A_log, Dv, hend, y + (size_t)bb * MT);
  }

  // 7) cross-branch combine
  combine_kernel<<<(int)(MT / 256), 256, 0, stream>>>(
      y, y + MT, y + 2 * MT, z, z + MT, z + 2 * MT, (float*)d_out, (int)MT);
}